// SPDRectified_79697413144764
// MI455X (gfx1250) — compile-verified
//
#include <hip/hip_runtime.h>

// CDNA5 (gfx1250) wave32 WMMA types
typedef float v2f __attribute__((ext_vector_type(2)));
typedef float v8f __attribute__((ext_vector_type(8)));

#define MATN   64
#define LS     68          // padded LDS row stride (floats): 16B-aligned rows, conflict-free column reads
#define NITER  30          // Newton-Schulz sign iterations
#define EPSV   1e-4f

// One 16x16 output tile accumulated over K=64 with v_wmma_f32_16x16x4_f32.
// Fragment layouts per CDNA5 ISA 7.12.2:
//   A (16x4 f32):  lane = 16*half + m;  v0 = A[m][k0+2*half],  v1 = A[m][k0+2*half+1]
//   B (4x16 f32):  vj : lanes0-15 -> row K=k0+j, lanes16-31 -> row K=k0+j+2
//   C/D (16x16):   VGPR r : lanes0-15 -> M=r, lanes16-31 -> M=r+8
__device__ __forceinline__ void mm_tile(const float* __restrict__ L,
                                        const float* __restrict__ R,
                                        int ti, int tj, int half, int m, v8f& acc)
{
    const int r0 = ti * 16;
    const int c0 = tj * 16;
    #pragma unroll
    for (int kb = 0; kb < 16; ++kb) {
        const int k0 = kb * 4;
        v2f a, b;
        a[0] = L[(r0 + m) * LS + k0 + 2 * half];
        a[1] = L[(r0 + m) * LS + k0 + 2 * half + 1];
        b[0] = R[(k0 + 2 * half + 0) * LS + c0 + m];
        b[1] = R[(k0 + 2 * half + 1) * LS + c0 + m];
        acc = __builtin_amdgcn_wmma_f32_16x16x4_f32(
            /*neg_a=*/false, a, /*neg_b=*/false, b,
            /*c_mod=*/(short)0, acc, /*reuse_a=*/false, /*reuse_b=*/false);
    }
}

__device__ __forceinline__ void store_tile(float* __restrict__ T,
                                           int ti, int tj, int half, int m, const v8f& acc)
{
    #pragma unroll
    for (int r = 0; r < 8; ++r)
        T[(ti * 16 + r + 8 * half) * LS + tj * 16 + m] = acc[r];
}

__global__ __launch_bounds__(256) void spd_rectify_kernel(const float* __restrict__ in,
                                                          float* __restrict__ out,
                                                          int batch)
{
    __shared__ float sB[MATN * LS];   // B = A - eps*I   (kept for final |B| = sign(B)*B)
    __shared__ float sX[MATN * LS];   // Newton-Schulz iterate
    __shared__ float sY[MATN * LS];   // temp = X*X
    __shared__ float red[256];

    const int t    = threadIdx.x;
    const int lane = t & 31;
    const int wave = t >> 5;
    const int half = lane >> 4;
    const int m    = lane & 15;
    const int bid  = blockIdx.x;
    if (bid >= batch) return;

    const float* A = in + (size_t)bid * (MATN * MATN);

    // hint next lines toward GL2 (emits global_prefetch_b8 on gfx1250)
    __builtin_prefetch(A + t * 16, 0, 1);

    // ---- stage B = A - eps*I into LDS (float4 global loads -> ds_store_b128) ----
    const float4* A4 = (const float4*)A;
    #pragma unroll
    for (int i = 0; i < 4; ++i) {
        const int vi = t + 256 * i;        // 0..1023 float4 chunks
        float4 v = A4[vi];
        const int r = vi >> 4;             // row
        const int c = (vi & 15) * 4;       // starting col
        if (r == c + 0) v.x -= EPSV;
        if (r == c + 1) v.y -= EPSV;
        if (r == c + 2) v.z -= EPSV;
        if (r == c + 3) v.w -= EPSV;
        *(float4*)&sB[r * LS + c] = v;
    }
    __syncthreads();

    // ---- Frobenius norm (deterministic tree reduction) ----
    float ss = 0.0f;
    #pragma unroll
    for (int i = 0; i < 16; ++i) {
        const int e = t + 256 * i;
        const float v = sB[(e >> 6) * LS + (e & 63)];
        ss += v * v;
    }
    red[t] = ss;
    __syncthreads();
    #pragma unroll
    for (int s = 128; s > 0; s >>= 1) {
        if (t < s) red[t] += red[t + s];
        __syncthreads();
    }
    const float scale = rsqrtf(red[0] + 1e-30f);   // ||X0||_2 <= ||X0||_F = 1

    // ---- X0 = B * scale ----
    #pragma unroll
    for (int i = 0; i < 16; ++i) {
        const int e = t + 256 * i;
        const int idx = (e >> 6) * LS + (e & 63);
        sX[idx] = sB[idx] * scale;
    }
    __syncthreads();

    // this wave's two 16x16 output tiles
    const int tile0 = wave * 2, tile1 = wave * 2 + 1;
    const int ti0 = tile0 >> 2, tj0 = tile0 & 3;
    const int ti1 = tile1 >> 2, tj1 = tile1 & 3;

    // ---- Newton-Schulz: X <- 1.5 X - 0.5 X^3  (sign(B)) ----
    for (int it = 0; it < NITER; ++it) {
        v8f acc0 = {}, acc1 = {};
        mm_tile(sX, sX, ti0, tj0, half, m, acc0);       // Y = X*X
        mm_tile(sX, sX, ti1, tj1, half, m, acc1);
        store_tile(sY, ti0, tj0, half, m, acc0);
        store_tile(sY, ti1, tj1, half, m, acc1);
        __syncthreads();

        v8f z0 = {}, z1 = {};
        mm_tile(sX, sY, ti0, tj0, half, m, z0);         // Z = X*Y = X^3
        mm_tile(sX, sY, ti1, tj1, half, m, z1);

        float x0[8], x1[8];
        #pragma unroll
        for (int r = 0; r < 8; ++r) {
            x0[r] = sX[(ti0 * 16 + r + 8 * half) * LS + tj0 * 16 + m];
            x1[r] = sX[(ti1 * 16 + r + 8 * half) * LS + tj1 * 16 + m];
        }
        __syncthreads();                                 // all reads of sX done
        #pragma unroll
        for (int r = 0; r < 8; ++r) {
            sX[(ti0 * 16 + r + 8 * half) * LS + tj0 * 16 + m] = 1.5f * x0[r] - 0.5f * z0[r];
            sX[(ti1 * 16 + r + 8 * half) * LS + tj1 * 16 + m] = 1.5f * x1[r] - 0.5f * z1[r];
        }
        __syncthreads();
    }

    // ---- out = 0.5*(B + sign(B)*B) + eps*I ----
    v8f w0 = {}, w1 = {};
    mm_tile(sX, sB, ti0, tj0, half, m, w0);
    mm_tile(sX, sB, ti1, tj1, half, m, w1);

    float* O = out + (size_t)bid * (MATN * MATN);
    #pragma unroll
    for (int r = 0; r < 8; ++r) {
        {
            const int row = ti0 * 16 + r + 8 * half, col = tj0 * 16 + m;
            float o = 0.5f * (sB[row * LS + col] + w0[r]);
            if (row == col) o += EPSV;
            O[row * MATN + col] = o;
        }
        {
            const int row = ti1 * 16 + r + 8 * half, col = tj1 * 16 + m;
            float o = 0.5f * (sB[row * LS + col] + w1[r]);
            if (row == col) o += EPSV;
            O[row * MATN + col] = o;
        }
    }
}

extern "C" void kernel_launch(void* const* d_in, const int* in_sizes, int n_in,
                              void* d_out, int out_size, void* d_ws, size_t ws_size,
                              hipStream_t stream)
{
    const float* in = (const float*)d_in[0];
    float* out = (float*)d_out;
    const int batch = in_sizes[0] / (MATN * MATN);   // 8192
    spd_rectify_kernel<<<batch, 256, 0, stream>>>(in, out, batch);
}